// ImageSeriesInterpolator_42855183679990
// MI455X (gfx1250) — compile-verified
//
#include <hip/hip_runtime.h>
#include <hip/hip_bf16.h>

// Problem constants from the reference: B=8, T=30, C=4, H=W=256.
#define T_STEPS 30
#define CH      4
#define HW      65536                    // 256*256, one image plane
#define PLANE_T ((size_t)CH * HW)        // elements per (b,t) image slab

// One thread per (b, h, w) pixel. Lanes cover consecutive w -> coalesced
// 128B transactions per wave32. Memory-bound: NT hints keep the single-use
// streams from thrashing L2; prefetch warms the visible planes after the
// batched mask scan.
__global__ __launch_bounds__(256) void ImageSeriesInterpolator_kernel(
    const float* __restrict__ images,   // [B,T,C,H,W]
    const float* __restrict__ mask,     // [B,T,1,H,W]
    const float* __restrict__ days,     // [B,T]
    float* __restrict__ out,            // [B,T,C,H,W]
    int total)                          // B*H*W
{
    const int tid = blockIdx.x * blockDim.x + threadIdx.x;
    if (tid >= total) return;

    const int b   = tid >> 16;          // HW == 2^16
    const int pix = tid & (HW - 1);

    const float* maskb = mask   + (size_t)b * T_STEPS * HW + pix;
    const float* imgb  = images + (size_t)b * T_STEPS * PLANE_T + pix;
    float*       outb  = out    + (size_t)b * T_STEPS * PLANE_T + pix;
    const float* daysb = days   + b * T_STEPS;   // tiny, reused -> regular temporal

    // ---- Pass 1a: batched mask loads (no branches between loads so the
    //      compiler can issue all 30 NT loads as one deep clause) ----
    float m[T_STEPS];
    #pragma unroll
    for (int t = 0; t < T_STEPS; ++t)
        m[t] = __builtin_nontemporal_load(maskb + (size_t)t * HW);

    // ---- Pass 1b: branchless visibility bitmask (bit t set => visible) ----
    unsigned int vis = 0u;
    #pragma unroll
    for (int t = 0; t < T_STEPS; ++t)
        vis |= (m[t] == 0.0f ? 1u : 0u) << t;

    // ---- Pass 1c: prefetch the planes the copy path will read shortly.
    //      Per-lane addresses; loop runs max-popcount times per wave. ----
    {
        unsigned int pb = vis;
        while (pb) {
            const int t = __builtin_ctz(pb);
            pb &= pb - 1u;
            __builtin_prefetch(imgb + (size_t)t * PLANE_T, 0, 1);
        }
    }

    // ---- Pass 2: walk time, carrying last-visible values in registers ----
    float lastv[CH] = {0.f, 0.f, 0.f, 0.f};
    float nextv[CH] = {0.f, 0.f, 0.f, 0.f};
    float d_last = 0.0f, d_next = 0.0f;
    int   lv = -1;
    int   nv = T_STEPS;
    bool  nv_stale = true;
    const float nanv = __builtin_nanf("");

    #pragma unroll 1
    for (int t = 0; t < T_STEPS; ++t) {
        const size_t base = (size_t)t * PLANE_T;
        if ((vis >> t) & 1u) {
            // Visible: straight copy; cache as img_last for the next cloudy run.
            float v0 = __builtin_nontemporal_load(imgb + base + 0 * (size_t)HW);
            float v1 = __builtin_nontemporal_load(imgb + base + 1 * (size_t)HW);
            float v2 = __builtin_nontemporal_load(imgb + base + 2 * (size_t)HW);
            float v3 = __builtin_nontemporal_load(imgb + base + 3 * (size_t)HW);
            __builtin_nontemporal_store(v0, outb + base + 0 * (size_t)HW);
            __builtin_nontemporal_store(v1, outb + base + 1 * (size_t)HW);
            __builtin_nontemporal_store(v2, outb + base + 2 * (size_t)HW);
            __builtin_nontemporal_store(v3, outb + base + 3 * (size_t)HW);
            lastv[0] = v0; lastv[1] = v1; lastv[2] = v2; lastv[3] = v3;
            lv = t;
            d_last = daysb[t];
            nv_stale = true;   // next-visible must be recomputed past this frame
        } else {
            if (nv_stale) {
                const unsigned int hi = vis >> t;     // t < 30 < 32
                if (hi) {
                    nv = t + __builtin_ctz(hi);
                    const size_t nb = (size_t)nv * PLANE_T;
                    nextv[0] = __builtin_nontemporal_load(imgb + nb + 0 * (size_t)HW);
                    nextv[1] = __builtin_nontemporal_load(imgb + nb + 1 * (size_t)HW);
                    nextv[2] = __builtin_nontemporal_load(imgb + nb + 2 * (size_t)HW);
                    nextv[3] = __builtin_nontemporal_load(imgb + nb + 3 * (size_t)HW);
                    d_next = daysb[nv];
                } else {
                    nv = T_STEPS;
                }
                nv_stale = false;
            }
            const bool has_l = (lv >= 0);
            const bool has_n = (nv < T_STEPS);
            const bool both  = has_l && has_n;

            // frac is only meaningful when both endpoints exist; otherwise the
            // lerp result is select-discarded (operands are defined -> no UB).
            const float d_t  = daysb[t];
            float denom = d_next - d_last;
            denom = (denom == 0.0f) ? 1.0f : denom;
            const float frac = (d_t - d_last) / denom;

            #pragma unroll
            for (int c = 0; c < CH; ++c) {
                const float lerp = __builtin_fmaf(frac, nextv[c] - lastv[c], lastv[c]);
                const float fallback = has_l ? lastv[c] : (has_n ? nextv[c] : nanv);
                const float r = both ? lerp : fallback;
                __builtin_nontemporal_store(r, outb + base + (size_t)c * HW);
            }
        }
    }
}

extern "C" void kernel_launch(void* const* d_in, const int* in_sizes, int n_in,
                              void* d_out, int out_size, void* d_ws, size_t ws_size,
                              hipStream_t stream) {
    const float* images = (const float*)d_in[0];   // [B,T,C,H,W] fp32
    const float* mask   = (const float*)d_in[1];   // [B,T,1,H,W] fp32
    const float* days   = (const float*)d_in[2];   // [B,T]       fp32
    float* out = (float*)d_out;

    const int B     = in_sizes[2] / T_STEPS;       // 240 / 30 = 8
    const int total = B * HW;                      // B*H*W = 524288 threads

    const int block = 256;                         // 8 wave32 per workgroup
    const int grid  = (total + block - 1) / block; // 2048 workgroups

    ImageSeriesInterpolator_kernel<<<grid, block, 0, stream>>>(
        images, mask, days, out, total);
}